// Codebook_22986664968251
// MI455X (gfx1250) — compile-verified
//
#include <hip/hip_runtime.h>
#include <hip/hip_bf16.h>
#include <math.h>

// ---------- problem constants ----------
#define N_TOKENS  131072      // 16 * 8192
#define EMBED_DIM 256
#define PROJ_DIM  32
#define N_CODES   1024
#define TILE_M    128         // tokens per workgroup in WMMA kernels

typedef __attribute__((ext_vector_type(2))) float v2f;
typedef __attribute__((ext_vector_type(8))) float v8f;

// D = A(16x4,f32) * B(4x16,f32) + C(16x16,f32)
__device__ __forceinline__ v8f wmma_f32_4(v2f a, v2f b, v8f c) {
  return __builtin_amdgcn_wmma_f32_16x16x4_f32(
      /*neg_a=*/false, a, /*neg_b=*/false, b,
      /*c_mod=*/(short)0, c, /*reuse_a=*/false, /*reuse_b=*/false);
}

__device__ __forceinline__ float wave_sum(float v) {
#pragma unroll
  for (int m = 16; m >= 1; m >>= 1) v += __shfl_xor(v, m, 32);
  return v;
}

// ---------- K0: zero accumulators in workspace ----------
__global__ void kzero(int* __restrict__ hist, float* __restrict__ sumsq) {
  int i = blockIdx.x * blockDim.x + threadIdx.x;
  if (i < N_CODES) hist[i] = 0;
  if (i == 0) *sumsq = 0.0f;
}

// ---------- K1: normalize codebook rows (one wave per row) ----------
__global__ __launch_bounds__(256) void kcbnorm(const float* __restrict__ cb,
                                               float* __restrict__ cbn) {
  const int lane = threadIdx.x & 31;
  const int wave = threadIdx.x >> 5;
  const int row  = blockIdx.x * 8 + wave;
  float v  = cb[row * PROJ_DIM + lane];
  float ss = wave_sum(v * v);
  float dn = fmaxf(sqrtf(ss), 1e-6f);
  cbn[row * PROJ_DIM + lane] = v / dn;
}

// ---------- K2: z_proj = normalize(z @ W_in), one wave per token ----------
__global__ __launch_bounds__(256) void kproj(const float* __restrict__ z,
                                             const float* __restrict__ Win,
                                             float* __restrict__ zproj) {
  const int lane = threadIdx.x & 31;
  const int wave = threadIdx.x >> 5;
  const long tok = (long)blockIdx.x * 8 + wave;
  const float* zr = z + tok * EMBED_DIM;
  // stream hint for the next block's row this wave position would read
  __builtin_prefetch(zr + 8 * EMBED_DIM, 0, 0);   // global_prefetch_b8

  // two b128 loads cover the whole 256-f32 row across the wave
  const float4 c0 = ((const float4*)zr)[lane];        // k = 4*lane   .. +3
  const float4 c1 = ((const float4*)zr)[32 + lane];   // k = 128+4*lane .. +3

  float acc = 0.0f;
#pragma unroll
  for (int j = 0; j < 32; ++j) {
    const int k = 4 * j;
    float z0 = __shfl(c0.x, j, 32), z1 = __shfl(c0.y, j, 32);
    float z2 = __shfl(c0.z, j, 32), z3 = __shfl(c0.w, j, 32);
    acc = fmaf(z0, Win[(k + 0) * PROJ_DIM + lane], acc);
    acc = fmaf(z1, Win[(k + 1) * PROJ_DIM + lane], acc);
    acc = fmaf(z2, Win[(k + 2) * PROJ_DIM + lane], acc);
    acc = fmaf(z3, Win[(k + 3) * PROJ_DIM + lane], acc);
  }
#pragma unroll
  for (int j = 0; j < 32; ++j) {
    const int k = 128 + 4 * j;
    float z0 = __shfl(c1.x, j, 32), z1 = __shfl(c1.y, j, 32);
    float z2 = __shfl(c1.z, j, 32), z3 = __shfl(c1.w, j, 32);
    acc = fmaf(z0, Win[(k + 0) * PROJ_DIM + lane], acc);
    acc = fmaf(z1, Win[(k + 1) * PROJ_DIM + lane], acc);
    acc = fmaf(z2, Win[(k + 2) * PROJ_DIM + lane], acc);
    acc = fmaf(z3, Win[(k + 3) * PROJ_DIM + lane], acc);
  }
  float ss = wave_sum(acc * acc);
  float dn = fmaxf(sqrtf(ss), 1e-6f);
  zproj[tok * PROJ_DIM + lane] = acc / dn;
}

// ---------- K3: distances via WMMA + argmin + histogram ----------
// scores = z_proj_tile[128x32] @ cbn^T[32x1024]; argmin(2-2*dot) == argmax(dot)
// Whole normalized codebook resident in LDS (128 KB) -> one barrier total.
__global__ __launch_bounds__(256) void kdist(const float* __restrict__ zproj,
                                             const float* __restrict__ cbn,
                                             int*   __restrict__ idx_i,
                                             float* __restrict__ idx_f,
                                             int*   __restrict__ hist) {
  __shared__ float sP[TILE_M * PROJ_DIM];    // 16 KB
  __shared__ float sC[N_CODES * PROJ_DIM];   // 128 KB
  const int tid  = threadIdx.x;
  const int wave = tid >> 5;
  const int lane = tid & 31;
  const int half = lane >> 4;
  const int l16  = lane & 15;
  const long tokbase = (long)blockIdx.x * TILE_M;

  {
    const float4* src = (const float4*)(zproj + tokbase * PROJ_DIM);
    float4* dst = (float4*)sP;
    for (int e = tid; e < TILE_M * PROJ_DIM / 4; e += 256) dst[e] = src[e];
    const float4* csrc = (const float4*)cbn;
    float4* cdst = (float4*)sC;
    for (int e = tid; e < N_CODES * PROJ_DIM / 4; e += 256) cdst[e] = csrc[e];
  }
  __syncthreads();

  // A fragments for this wave's 16 rows (16x32 tile -> 8 K-steps of 16x4)
  v2f a[8];
  const int mrow = wave * 16 + l16;
#pragma unroll
  for (int ks = 0; ks < 8; ++ks) {
    const int k = ks * 4 + half * 2;
    a[ks].x = sP[mrow * PROJ_DIM + k];
    a[ks].y = sP[mrow * PROJ_DIM + k + 1];
  }

  float bestv[8];
  int   besti[8];
#pragma unroll
  for (int r = 0; r < 8; ++r) { bestv[r] = -3.4e38f; besti[r] = 0; }

#pragma unroll 4
  for (int nt = 0; nt < N_CODES / 16; ++nt) {   // 64 N-tiles of 16 codes
    const int nrow = nt * 16 + l16;             // code row (B = cbn^T)
    const float* bp = &sC[nrow * PROJ_DIM + half * 2];
    v2f b[8];                                   // preload all B fragments
#pragma unroll
    for (int ks = 0; ks < 8; ++ks) { b[ks].x = bp[ks * 4]; b[ks].y = bp[ks * 4 + 1]; }
    v8f acc = {0.f, 0.f, 0.f, 0.f, 0.f, 0.f, 0.f, 0.f};
#pragma unroll
    for (int ks = 0; ks < 8; ++ks) acc = wmma_f32_4(a[ks], b[ks], acc);
    const int nidx = nt * 16 + l16;             // column this lane owns
#pragma unroll
    for (int r = 0; r < 8; ++r) {
      const float v = acc[r];
      if (v > bestv[r]) { bestv[r] = v; besti[r] = nidx; }  // keep first on tie
    }
  }

  // argmax reduction across the 16 lanes that share each C row
#pragma unroll
  for (int r = 0; r < 8; ++r) {
    float v  = bestv[r];
    int   ix = besti[r];
#pragma unroll
    for (int m = 8; m >= 1; m >>= 1) {
      const float ov  = __shfl_xor(v, m, 32);
      const int   oix = __shfl_xor(ix, m, 32);
      if (ov > v || (ov == v && oix < ix)) { v = ov; ix = oix; }
    }
    if (l16 == 0) {
      const long tok = tokbase + wave * 16 + half * 8 + r; // C-row mapping
      idx_i[tok] = ix;
      idx_f[tok] = (float)ix;
      atomicAdd(&hist[ix], 1);
    }
  }
}

// ---------- K4: gather cb[idx], WMMA output GEMM, fused loss ----------
__global__ __launch_bounds__(256) void kout(const float* __restrict__ zproj,
                                            const float* __restrict__ cbn,
                                            const int*   __restrict__ idx_i,
                                            const float* __restrict__ Wout,
                                            float* __restrict__ emb,
                                            float* __restrict__ sumsq) {
  __shared__ float sW[PROJ_DIM * EMBED_DIM];  // 32 KB
  __shared__ float sQ[TILE_M * PROJ_DIM];     // 16 KB
  __shared__ int   sIdx[TILE_M];
  const int tid  = threadIdx.x;
  const int wave = tid >> 5;
  const int lane = tid & 31;
  const int half = lane >> 4;
  const int l16  = lane & 15;
  const long tokbase = (long)blockIdx.x * TILE_M;

  {
    const float4* wsrc = (const float4*)Wout;
    float4* wdst = (float4*)sW;
    for (int e = tid; e < PROJ_DIM * EMBED_DIM / 4; e += 256) wdst[e] = wsrc[e];
  }
  if (tid < TILE_M) sIdx[tid] = idx_i[tokbase + tid];
  __syncthreads();
  {
    const float4* qsrc = (const float4*)cbn;
    float4* qdst = (float4*)sQ;
    for (int e = tid; e < TILE_M * PROJ_DIM / 4; e += 256) {
      const int row = e >> 3, col = e & 7;      // 8 float4 per 32-f32 row
      qdst[e] = qsrc[sIdx[row] * (PROJ_DIM / 4) + col];
    }
  }
  __syncthreads();

  // fused sum ||z_proj - quantized||^2 (vectorized)
  float part = 0.0f;
  {
    const float4* psrc = (const float4*)(zproj + tokbase * PROJ_DIM);
    const float4* qsrc = (const float4*)sQ;
    for (int e = tid; e < TILE_M * PROJ_DIM / 4; e += 256) {
      const float4 p = psrc[e], q = qsrc[e];
      const float d0 = p.x - q.x, d1 = p.y - q.y, d2 = p.z - q.z, d3 = p.w - q.w;
      part = fmaf(d0, d0, part); part = fmaf(d1, d1, part);
      part = fmaf(d2, d2, part); part = fmaf(d3, d3, part);
    }
  }
  part = wave_sum(part);
  if (lane == 0) atomicAdd(sumsq, part);

  // A fragments from quantized tile
  v2f a[8];
  const int mrow = wave * 16 + l16;
#pragma unroll
  for (int ks = 0; ks < 8; ++ks) {
    const int k = ks * 4 + half * 2;
    a[ks].x = sQ[mrow * PROJ_DIM + k];
    a[ks].y = sQ[mrow * PROJ_DIM + k + 1];
  }
  // emb_tile[16x256] = Q[16x32] @ W_out[32x256]
#pragma unroll 2
  for (int nt = 0; nt < EMBED_DIM / 16; ++nt) {
    const int n = nt * 16 + l16;
    v2f b[8];                                   // preload all B fragments
#pragma unroll
    for (int ks = 0; ks < 8; ++ks) {
      const int k = ks * 4 + half * 2;
      b[ks].x = sW[k * EMBED_DIM + n];
      b[ks].y = sW[(k + 1) * EMBED_DIM + n];
    }
    v8f acc = {0.f, 0.f, 0.f, 0.f, 0.f, 0.f, 0.f, 0.f};
#pragma unroll
    for (int ks = 0; ks < 8; ++ks) acc = wmma_f32_4(a[ks], b[ks], acc);
#pragma unroll
    for (int r = 0; r < 8; ++r) {
      const long row = tokbase + wave * 16 + half * 8 + r;
      emb[row * EMBED_DIM + n] = acc[r];
    }
  }
}

// ---------- K5: finalize scalars ----------
__global__ __launch_bounds__(256) void kfinal(const int* __restrict__ hist,
                                              const float* __restrict__ sumsq,
                                              float* __restrict__ out_scalars) {
  __shared__ float red[8];
  const int tid = threadIdx.x, wave = tid >> 5, lane = tid & 31;
  float part = 0.0f;
  for (int i = tid; i < N_CODES; i += 256) {
    const float p = (float)hist[i] * (1.0f / (float)N_TOKENS);
    part += p * logf(p + 1e-10f);
  }
  part = wave_sum(part);
  if (lane == 0) red[wave] = part;
  __syncthreads();
  if (tid == 0) {
    float s = 0.0f;
#pragma unroll
    for (int w = 0; w < 8; ++w) s += red[w];
    const float perplexity = expf(-s);
    const float mean_half  = 0.5f * (*sumsq) / ((float)N_TOKENS * (float)PROJ_DIM);
    out_scalars[0] = 0.25f * mean_half;  // commitment_loss
    out_scalars[1] = mean_half;          // codebook_loss
    out_scalars[2] = perplexity;
  }
}

extern "C" void kernel_launch(void* const* d_in, const int* in_sizes, int n_in,
                              void* d_out, int out_size, void* d_ws, size_t ws_size,
                              hipStream_t stream) {
  const float* z    = (const float*)d_in[0];  // [16,8192,256]
  const float* cb   = (const float*)d_in[1];  // [1024,32]
  const float* Win  = (const float*)d_in[2];  // [256,32]
  const float* Wout = (const float*)d_in[3];  // [32,256]

  // d_out layout: embeddings | indices(float) | commitment | codebook | perplexity
  float* emb     = (float*)d_out;
  float* idx_f   = emb + (long)N_TOKENS * EMBED_DIM;
  float* scalars = idx_f + N_TOKENS;

  // workspace layout (bytes)
  char* ws = (char*)d_ws;
  float* cbn   = (float*)(ws);                       // 1024*32*4   = 128 KB
  float* zproj = (float*)(ws + (1 << 17));           // 131072*32*4 = 16 MB
  int*   idx_i = (int*)  (ws + (1 << 17) + (1 << 24));
  int*   hist  = (int*)  (ws + (1 << 17) + (1 << 24) + N_TOKENS * 4);
  float* sumsq = (float*)(ws + (1 << 17) + (1 << 24) + N_TOKENS * 4 + N_CODES * 4);

  kzero  <<<4, 256, 0, stream>>>(hist, sumsq);
  kcbnorm<<<N_CODES / 8, 256, 0, stream>>>(cb, cbn);
  kproj  <<<N_TOKENS / 8, 256, 0, stream>>>(z, Win, zproj);
  kdist  <<<N_TOKENS / TILE_M, 256, 0, stream>>>(zproj, cbn, idx_i, idx_f, hist);
  kout   <<<N_TOKENS / TILE_M, 256, 0, stream>>>(zproj, cbn, idx_i, Wout, emb, sumsq);
  kfinal <<<1, 256, 0, stream>>>(hist, sumsq, scalars);
}